// GCNN_343597384357
// MI455X (gfx1250) — compile-verified
//
#include <hip/hip_runtime.h>

// ---------------------------------------------------------------------------
// GraphConv x2 + sum-pool for MI455X (gfx1250, wave32).
//   N=50000 nodes, E=800000 edges, D=64 features, 64 graphs.
//   agg = scatter_add(x[src] -> dst);  h = relu(agg@W_rel + x@W_root + b)
//   GEMMs via V_WMMA_F32_16X16X4_F32 (fp32, matches reference numerics).
//   Activation tiles staged into LDS with GLOBAL_LOAD_ASYNC_TO_LDS_B128
//   (ASYNCcnt path) for full tiles; sync zero-padded fallback at the edge.
// ---------------------------------------------------------------------------

typedef float v2f __attribute__((ext_vector_type(2)));
typedef float v8f __attribute__((ext_vector_type(8)));

#define D          64
#define TS         68   // padded LDS row stride (floats): banks 4r+k, 16B aligned
#define NODE_TILE  64   // nodes per workgroup in the linear kernel

// Hardware float atomic add (L2 fadd path); fallback to generic atomicAdd.
__device__ __forceinline__ void atomic_fadd(float* p, float v) {
#if __has_builtin(__builtin_amdgcn_global_atomic_fadd_f32)
  typedef __attribute__((address_space(1))) float gfloat;
  __builtin_amdgcn_global_atomic_fadd_f32((gfloat*)(unsigned long long)p, v);
#else
  atomicAdd(p, v);
#endif
}

// Per-lane async DMA: 16 bytes global -> LDS, tracked by ASYNCcnt.
// lds_off is the wave-relative LDS byte offset (low 32 bits of the flat
// LDS address -- hardware truncates the same way).
__device__ __forceinline__ void async_copy_b128(unsigned lds_off, const float* gp) {
  asm volatile("global_load_async_to_lds_b128 %0, %1, off"
               :: "v"(lds_off), "v"(gp) : "memory");
}

__device__ __forceinline__ void wait_asynccnt0() {
#if __has_builtin(__builtin_amdgcn_s_wait_asynccnt)
  __builtin_amdgcn_s_wait_asynccnt(0);
#else
  asm volatile("s_wait_asynccnt 0x0" ::: "memory");
#endif
}

// ---------------------------------------------------------------------------
__global__ void zero_kernel(float* __restrict__ p, long long n) {
  long long i = (long long)blockIdx.x * blockDim.x + threadIdx.x;
  if (i < n) p[i] = 0.0f;
}

// ---------------------------------------------------------------------------
// Edge scatter: 16 threads per edge, each handling a float4 chunk of the
// 64-float feature row.  Gather is coalesced per-edge; adds go through the
// f32 atomic-add units (whole working set fits in the 192MB L2).
__global__ void edge_scatter_kernel(const float* __restrict__ feat,
                                    const int* __restrict__ src,
                                    const int* __restrict__ dst,
                                    float* __restrict__ agg, int E) {
  long long t = (long long)blockIdx.x * blockDim.x + threadIdx.x;
  if (t >= (long long)E * 16) return;
  int e = (int)(t >> 4);
  int c = ((int)t & 15) * 4;
  int s = src[e];
  int d = dst[e];
  const float4 v = *(const float4*)(feat + (long long)s * D + c);
  float* base = agg + (long long)d * D + c;
  atomic_fadd(base + 0, v.x);
  atomic_fadd(base + 1, v.y);
  atomic_fadd(base + 2, v.z);
  atomic_fadd(base + 3, v.w);
}

// ---------------------------------------------------------------------------
// Fused dual-GEMM + bias + ReLU (+ optional fused graph sum-pool):
//   out = relu(Aagg @ Wrel + Aroot @ Wroot + bias)
// One workgroup = 64 node rows.  8 waves, each owning two 16x16 output tiles
// that share the same column tile (B fragment loaded once).  K accumulated in
// 16 steps of V_WMMA_F32_16X16X4_F32 per weight matrix.
__global__ __launch_bounds__(256)
void linear_relu_wmma_kernel(const float* __restrict__ Aagg,
                             const float* __restrict__ Aroot,
                             const float* __restrict__ Wrel,
                             const float* __restrict__ Wroot,
                             const float* __restrict__ bias,
                             float* __restrict__ out,       // may be null
                             const int* __restrict__ batch, // used iff pool
                             float* __restrict__ pool,      // may be null
                             int N) {
  __shared__ float sWrel[D * TS];   // transposed: sWrel[col*TS + k] = Wrel[k][col]
  __shared__ float sWroot[D * TS];
  __shared__ float sT[NODE_TILE * TS];

  const int tid = threadIdx.x;
  const int blockBase = blockIdx.x * NODE_TILE;
  const bool fullTile = (blockBase + NODE_TILE) <= N;   // uniform per block

  // Kick off async DMA of the agg tile into LDS (overlaps weight staging).
  if (fullTile) {
#pragma unroll
    for (int ii = 0; ii < 4; ii++) {
      int idx = tid + ii * 256;            // 1024 x 16B chunks
      int r = idx >> 4;
      int c = (idx & 15) * 4;
      unsigned lds = (unsigned)(unsigned long long)(const void*)&sT[r * TS + c];
      async_copy_b128(lds, Aagg + (long long)(blockBase + r) * D + c);
    }
  } else {
#pragma unroll
    for (int ii = 0; ii < 4; ii++) {
      int idx = tid + ii * 256;
      int r = idx >> 4;
      int c = (idx & 15) * 4;
      int node = blockBase + r;
      float4 v = make_float4(0.f, 0.f, 0.f, 0.f);
      if (node < N) v = *(const float4*)(Aagg + (long long)node * D + c);
      *(float4*)&sT[r * TS + c] = v;
    }
  }

  // Stage both weight matrices transposed (coalesced global read).
  for (int i = tid; i < D * D; i += 256) {
    int k = i >> 6, c = i & 63;
    sWrel[c * TS + k]  = Wrel[i];
    sWroot[c * TS + k] = Wroot[i];
  }

  if (fullTile) wait_asynccnt0();
  __syncthreads();

  const int wave = tid >> 5;
  const int lane = tid & 31;
  const int half = lane >> 4;   // K sub-split across half-waves
  const int lr   = lane & 15;
  const int colTile  = wave & 3;   // 0..3 -> output cols [16*colTile, +16)
  const int rowTile0 = wave >> 2;  // 0..1 ; second tile is rowTile0+2

  v8f acc0 = {};
  v8f acc1 = {};

  const int aRow0 = (rowTile0 * 16 + lr) * TS;
  const int aRow1 = ((rowTile0 + 2) * 16 + lr) * TS;
  const int bRow  = (colTile * 16 + lr) * TS;
  const int kOff  = 2 * half;   // lanes 16..31 carry K+2,K+3

  // Pass 1: agg @ Wrel
#pragma unroll
  for (int k = 0; k < D; k += 4) {
    v2f a0 = *(const v2f*)(sT + aRow0 + k + kOff);
    v2f a1 = *(const v2f*)(sT + aRow1 + k + kOff);
    v2f b  = *(const v2f*)(sWrel + bRow + k + kOff);
    acc0 = __builtin_amdgcn_wmma_f32_16x16x4_f32(false, a0, false, b,
                                                 (short)0, acc0, false, false);
    acc1 = __builtin_amdgcn_wmma_f32_16x16x4_f32(false, a1, false, b,
                                                 (short)0, acc1, false, false);
  }

  __syncthreads();   // everyone done reading sT

  // Re-stage sT with the root-input tile (async DMA on full tiles).
  if (fullTile) {
#pragma unroll
    for (int ii = 0; ii < 4; ii++) {
      int idx = tid + ii * 256;
      int r = idx >> 4;
      int c = (idx & 15) * 4;
      unsigned lds = (unsigned)(unsigned long long)(const void*)&sT[r * TS + c];
      async_copy_b128(lds, Aroot + (long long)(blockBase + r) * D + c);
    }
    wait_asynccnt0();
  } else {
#pragma unroll
    for (int ii = 0; ii < 4; ii++) {
      int idx = tid + ii * 256;
      int r = idx >> 4;
      int c = (idx & 15) * 4;
      int node = blockBase + r;
      float4 v = make_float4(0.f, 0.f, 0.f, 0.f);
      if (node < N) v = *(const float4*)(Aroot + (long long)node * D + c);
      *(float4*)&sT[r * TS + c] = v;
    }
  }
  __syncthreads();

  // Pass 2: += root @ Wroot
#pragma unroll
  for (int k = 0; k < D; k += 4) {
    v2f a0 = *(const v2f*)(sT + aRow0 + k + kOff);
    v2f a1 = *(const v2f*)(sT + aRow1 + k + kOff);
    v2f b  = *(const v2f*)(sWroot + bRow + k + kOff);
    acc0 = __builtin_amdgcn_wmma_f32_16x16x4_f32(false, a0, false, b,
                                                 (short)0, acc0, false, false);
    acc1 = __builtin_amdgcn_wmma_f32_16x16x4_f32(false, a1, false, b,
                                                 (short)0, acc1, false, false);
  }

  // Epilogue: bias + ReLU, then store and/or fused pooling.
  const int col = colTile * 16 + lr;
  const float bv = bias[col];

#pragma unroll
  for (int i = 0; i < 8; i++) {
    int m = i + 8 * half;  // C/D layout: VGPR i holds M=i (lanes<16), M=i+8 (lanes>=16)
    int node0 = blockBase + rowTile0 * 16 + m;
    float v0 = fmaxf(acc0[i] + bv, 0.0f);
    if (node0 < N) {
      if (out)  out[(long long)node0 * D + col] = v0;
      if (pool) atomic_fadd(pool + batch[node0] * D + col, v0);
    }
    int node1 = blockBase + (rowTile0 + 2) * 16 + m;
    float v1 = fmaxf(acc1[i] + bv, 0.0f);
    if (node1 < N) {
      if (out)  out[(long long)node1 * D + col] = v1;
      if (pool) atomic_fadd(pool + batch[node1] * D + col, v1);
    }
  }
}

// ---------------------------------------------------------------------------
extern "C" void kernel_launch(void* const* d_in, const int* in_sizes, int n_in,
                              void* d_out, int out_size, void* d_ws, size_t ws_size,
                              hipStream_t stream) {
  const float* x       = (const float*)d_in[0];
  const int*   edge    = (const int*)d_in[1];   // [2, E] flat
  const int*   batch   = (const int*)d_in[2];   // [N]
  const float* W1_rel  = (const float*)d_in[3];
  const float* b1      = (const float*)d_in[4];
  const float* W1_root = (const float*)d_in[5];
  const float* W2_rel  = (const float*)d_in[6];
  const float* b2      = (const float*)d_in[7];
  const float* W2_root = (const float*)d_in[8];
  float* out = (float*)d_out;

  const int N = in_sizes[0] / D;
  const int E = in_sizes[1] / 2;
  const int* src = edge;
  const int* dst = edge + E;

  float* agg = (float*)d_ws;                 // N*64 floats, reused per layer
  float* h1  = agg + (size_t)N * D;          // N*64 floats

  const long long nAgg = (long long)N * D;
  const int zgridAgg = (int)((nAgg + 255) / 256);
  const int zgridOut = (out_size + 255) / 256;

  const long long sThreads = (long long)E * 16;
  const int sgrid = (int)((sThreads + 255) / 256);

  const int lgrid = (N + NODE_TILE - 1) / NODE_TILE;

  // ---- Layer 1 ----
  zero_kernel<<<zgridAgg, 256, 0, stream>>>(agg, nAgg);
  zero_kernel<<<zgridOut, 256, 0, stream>>>(out, out_size);
  edge_scatter_kernel<<<sgrid, 256, 0, stream>>>(x, src, dst, agg, E);
  linear_relu_wmma_kernel<<<lgrid, 256, 0, stream>>>(
      agg, x, W1_rel, W1_root, b1, h1, nullptr, nullptr, N);

  // ---- Layer 2 (pooling fused into epilogue) ----
  zero_kernel<<<zgridAgg, 256, 0, stream>>>(agg, nAgg);
  edge_scatter_kernel<<<sgrid, 256, 0, stream>>>(h1, src, dst, agg, E);
  linear_relu_wmma_kernel<<<lgrid, 256, 0, stream>>>(
      agg, h1, W2_rel, W2_root, b2, nullptr, batch, out, N);
}